// LGNTDIC_89550068122385
// MI455X (gfx1250) — compile-verified
//
#include <hip/hip_runtime.h>
#include <hip/hip_bf16.h>

// ---------------------------------------------------------------------------
// LightGCN + TIDE loss on gfx1250 (wave32).
//  - Propagation: bandwidth-bound gather/scale/scatter with global_atomic_add_f32
//  - Scoring:     V_WMMA_F32_16X16X4_F32 chains computing diag(U * P^T)
// ---------------------------------------------------------------------------

typedef __attribute__((ext_vector_type(2))) float v2f;
typedef __attribute__((ext_vector_type(8))) float v8f;

__device__ __forceinline__ float logsigf(float x) {
    // log(sigmoid(x)) = min(x,0) - log1p(exp(-|x|))
    return fminf(x, 0.0f) - log1pf(expf(-fabsf(x)));
}
__device__ __forceinline__ float softplusf(float x) {
    // log1p(exp(x)) = max(x,0) + log1p(exp(-|x|))
    return fmaxf(x, 0.0f) + log1pf(expf(-fabsf(x)));
}

// --- degree accumulation ----------------------------------------------------
__global__ void k_degree(const int* __restrict__ src, const int* __restrict__ dst,
                         float* __restrict__ dout, float* __restrict__ din, int E) {
    int e = blockIdx.x * blockDim.x + threadIdx.x;
    if (e < E) {
        unsafeAtomicAdd(&dout[src[e]], 1.0f);
        unsafeAtomicAdd(&din[dst[e]], 1.0f);
    }
}

// --- rsqrt(max(deg,1)) in place --------------------------------------------
__global__ void k_rinv(float* __restrict__ a, float* __restrict__ b, int n) {
    int i = blockIdx.x * blockDim.x + threadIdx.x;
    if (i < n) {
        a[i] = rsqrtf(fmaxf(a[i], 1.0f));
        b[i] = rsqrtf(fmaxf(b[i], 1.0f));
    }
}

// --- per-edge symmetric normalization --------------------------------------
__global__ void k_norm(const int* __restrict__ src, const int* __restrict__ dst,
                       const float* __restrict__ ro, const float* __restrict__ ri,
                       float* __restrict__ norm, int E) {
    int e = blockIdx.x * blockDim.x + threadIdx.x;
    if (e < E) norm[e] = ro[src[e]] * ri[dst[e]];
}

// --- gather-scale-scatter: out[dst] += h[src] * norm  (16 threads/edge, f4) -
__global__ void k_scatter(const float* __restrict__ h, float* __restrict__ out,
                          const int* __restrict__ src, const int* __restrict__ dst,
                          const float* __restrict__ norm, int E) {
    unsigned long long tid = (unsigned long long)blockIdx.x * blockDim.x + threadIdx.x;
    unsigned int e = (unsigned int)(tid >> 4);
    unsigned int c = ((unsigned int)tid & 15u) << 2;   // float offset in row
    if (e >= (unsigned int)E) return;
    int s = src[e], d = dst[e];
    float w = norm[e];
    const float4 v = *reinterpret_cast<const float4*>(h + (size_t)s * 64 + c);
    float* o = out + (size_t)d * 64 + c;
    unsafeAtomicAdd(o + 0, v.x * w);
    unsafeAtomicAdd(o + 1, v.y * w);
    unsafeAtomicAdd(o + 2, v.z * w);
    unsafeAtomicAdd(o + 3, v.w * w);
}

// --- acc = e0 + h -----------------------------------------------------------
__global__ void k_add_init(const float* __restrict__ e0, const float* __restrict__ h,
                           float* __restrict__ acc, size_t n4) {
    size_t i = (size_t)blockIdx.x * blockDim.x + threadIdx.x;
    if (i < n4) {
        float4 a = reinterpret_cast<const float4*>(e0)[i];
        float4 b = reinterpret_cast<const float4*>(h)[i];
        float4 r = {a.x + b.x, a.y + b.y, a.z + b.z, a.w + b.w};
        reinterpret_cast<float4*>(acc)[i] = r;
    }
}

// --- acc += h ---------------------------------------------------------------
__global__ void k_add(float* __restrict__ acc, const float* __restrict__ h, size_t n4) {
    size_t i = (size_t)blockIdx.x * blockDim.x + threadIdx.x;
    if (i < n4) {
        float4 a = reinterpret_cast<float4*>(acc)[i];
        float4 b = reinterpret_cast<const float4*>(h)[i];
        float4 r = {a.x + b.x, a.y + b.y, a.z + b.z, a.w + b.w};
        reinterpret_cast<float4*>(acc)[i] = r;
    }
}

// --- scoring + loss: 16 triples per wave, diag(U P^T) via WMMA f32 16x16x4 --
__global__ void k_score(const float* __restrict__ fi, const float* __restrict__ fp,
                        const float* __restrict__ q, const float* __restrict__ bq,
                        const int* __restrict__ user, const int* __restrict__ itp,
                        const int* __restrict__ itn, const unsigned char* __restrict__ mask,
                        float* __restrict__ out, int B, int n_user) {
    const int lane = threadIdx.x & 31;
    const int wv   = threadIdx.x >> 5;
    const int grp  = blockIdx.x * (blockDim.x >> 5) + wv;
    const int base = grp * 16;
    if (base >= B) return;                    // wave-uniform: EXEC stays full below

    const int m = lane & 15;                  // triple handled by this lane (dup'd hi half)
    const int t = base + m;
    const size_t ur = (size_t)user[t] * 64;
    const size_t pr = (size_t)(itp[t] + n_user) * 64;
    const size_t nr = (size_t)(itn[t] + n_user) * 64;

    // 16x4 f32 A fragment and 4x16 f32 B fragment have identical per-lane
    // source layout: row = lane&15, k = k0 + 2*(lane>>4), two consecutive f32.
    const int khalf = (lane >> 4) << 1;

    v8f c_pi = {}; v8f c_ni = {}; v8f c_pp = {}; v8f c_np = {};
    #pragma unroll
    for (int k0 = 0; k0 < 64; k0 += 4) {
        const int kk = k0 + khalf;
        v2f a_ui = *reinterpret_cast<const v2f*>(fi + ur + kk);
        v2f b_pi = *reinterpret_cast<const v2f*>(fi + pr + kk);
        v2f b_ni = *reinterpret_cast<const v2f*>(fi + nr + kk);
        v2f a_up = *reinterpret_cast<const v2f*>(fp + ur + kk);
        v2f b_pp = *reinterpret_cast<const v2f*>(fp + pr + kk);
        v2f b_np = *reinterpret_cast<const v2f*>(fp + nr + kk);
        c_pi = __builtin_amdgcn_wmma_f32_16x16x4_f32(false, a_ui, false, b_pi, (short)0, c_pi, false, false);
        c_ni = __builtin_amdgcn_wmma_f32_16x16x4_f32(false, a_ui, false, b_ni, (short)0, c_ni, false, false);
        c_pp = __builtin_amdgcn_wmma_f32_16x16x4_f32(false, a_up, false, b_pp, (short)0, c_pp, false, false);
        c_np = __builtin_amdgcn_wmma_f32_16x16x4_f32(false, a_up, false, b_np, (short)0, c_np, false, false);
    }

    // Diagonal (m,m): VGPR m&7, lane m (m<8) or lane m+16 (m>=8).
    float d_pi = 0.f, d_ni = 0.f, d_pp = 0.f, d_np = 0.f;
    #pragma unroll
    for (int r = 0; r < 8; ++r) {
        float lo, hi;
        lo = __shfl(c_pi[r], r, 32); hi = __shfl(c_pi[r], r + 24, 32);
        if ((lane & 7) == r) d_pi = (lane & 8) ? hi : lo;
        lo = __shfl(c_ni[r], r, 32); hi = __shfl(c_ni[r], r + 24, 32);
        if ((lane & 7) == r) d_ni = (lane & 8) ? hi : lo;
        lo = __shfl(c_pp[r], r, 32); hi = __shfl(c_pp[r], r + 24, 32);
        if ((lane & 7) == r) d_pp = (lane & 8) ? hi : lo;
        lo = __shfl(c_np[r], r, 32); hi = __shfl(c_np[r], r + 24, 32);
        if ((lane & 7) == r) d_np = (lane & 8) ? hi : lo;
    }

    // f = acc/4 for both operands -> dot scales by 1/16
    const float scale = 1.0f / 16.0f;
    const float p_int = d_pi * scale, n_int = d_ni * scale;
    const float p_pop = d_pp * scale, n_pop = d_np * scale;

    const int ip = itp[t], inn = itn[t];
    const float mf  = mask[t] ? 1.0f : 0.0f;
    const float nmf = 1.0f - mf;
    const float pop_p = softplusf(q[ip]) + softplusf(bq[ip]);
    const float pop_n = softplusf(q[inn]) + softplusf(bq[inn]);

    const float l_int  = -mf * logsigf(p_int - n_int);
    const float l_pop  = -mf * logsigf(n_pop - p_pop) + nmf * logsigf(p_pop - n_pop);
    const float l_tide = -logsigf(tanhf(pop_p) * (p_int + p_pop) - tanhf(pop_n) * (n_int + n_pop));

    float ct = (0.1f * l_int + 0.1f * l_pop + 0.2f * l_tide) / (float)B;
    float v = (lane < 16) ? ct : 0.0f;        // upper half holds duplicates
    #pragma unroll
    for (int off = 16; off; off >>= 1) v += __shfl_down(v, off, 32);

    __shared__ float ssum[8];
    if (lane == 0) ssum[wv] = v;
    __syncthreads();
    if (threadIdx.x == 0) {
        float s = 0.0f;
        const int nw = blockDim.x >> 5;
        for (int i = 0; i < nw; ++i) s += ssum[i];
        unsafeAtomicAdd(out, s);
    }
}

extern "C" void kernel_launch(void* const* d_in, const int* in_sizes, int n_in,
                              void* d_out, int out_size, void* d_ws, size_t ws_size,
                              hipStream_t stream) {
    const float* emb_int = (const float*)d_in[0];
    const float* emb_pop = (const float*)d_in[1];
    const float* q    = (const float*)d_in[2];
    const float* bq   = (const float*)d_in[3];
    const int*   user = (const int*)d_in[4];
    const int*   itp  = (const int*)d_in[5];
    const int*   itn  = (const int*)d_in[6];
    const unsigned char* mask = (const unsigned char*)d_in[7];   // jax bool = 1 byte
    const int*   esrc = (const int*)d_in[8];
    const int*   edst = (const int*)d_in[9];

    const int D  = 64;
    const int N  = in_sizes[0] / D;
    const int NI = in_sizes[2];
    const int NU = N - NI;
    const int E  = in_sizes[8];
    const int B  = in_sizes[4];
    const size_t ND = (size_t)N * D;

    // Carve workspace (256B aligned): deg/rinv x2, norm[E], h0, h1, acc_int, acc_pop
    char* w = (char*)d_ws;
    auto carve = [&](size_t bytes) -> float* {
        float* p = (float*)w;
        w += (bytes + 255) & ~(size_t)255;
        return p;
    };
    float* deg_out = carve((size_t)N * 4);
    float* deg_in  = carve((size_t)N * 4);
    float* norm    = carve((size_t)E * 4);
    float* h0      = carve(ND * 4);
    float* h1      = carve(ND * 4);
    float* acc_i   = carve(ND * 4);
    float* acc_p   = carve(ND * 4);

    const int tb = 256;

    hipMemsetAsync(deg_out, 0, (size_t)N * 4, stream);
    hipMemsetAsync(deg_in,  0, (size_t)N * 4, stream);
    k_degree<<<(E + tb - 1) / tb, tb, 0, stream>>>(esrc, edst, deg_out, deg_in, E);
    k_rinv<<<(N + tb - 1) / tb, tb, 0, stream>>>(deg_out, deg_in, N);
    k_norm<<<(E + tb - 1) / tb, tb, 0, stream>>>(esrc, edst, deg_out, deg_in, norm, E);

    const float* embs[2] = {emb_int, emb_pop};
    float* accs[2] = {acc_i, acc_p};
    const unsigned int scatBlocks = (unsigned int)(((unsigned long long)E * 16 + tb - 1) / tb);
    const size_t n4 = ND / 4;
    const unsigned int addBlocks = (unsigned int)((n4 + tb - 1) / tb);

    for (int ei = 0; ei < 2; ++ei) {
        hipMemsetAsync(h0, 0, ND * 4, stream);
        k_scatter<<<scatBlocks, tb, 0, stream>>>(embs[ei], h0, esrc, edst, norm, E);
        k_add_init<<<addBlocks, tb, 0, stream>>>(embs[ei], h0, accs[ei], n4);

        hipMemsetAsync(h1, 0, ND * 4, stream);
        k_scatter<<<scatBlocks, tb, 0, stream>>>(h0, h1, esrc, edst, norm, E);
        k_add<<<addBlocks, tb, 0, stream>>>(accs[ei], h1, n4);

        hipMemsetAsync(h0, 0, ND * 4, stream);
        k_scatter<<<scatBlocks, tb, 0, stream>>>(h1, h0, esrc, edst, norm, E);
        k_add<<<addBlocks, tb, 0, stream>>>(accs[ei], h0, n4);
    }

    hipMemsetAsync(d_out, 0, sizeof(float), stream);
    const int groups = (B + 15) / 16;          // 16 triples per wave
    const int wavesPerBlock = 8;               // 256 threads
    const int sBlocks = (groups + wavesPerBlock - 1) / wavesPerBlock;
    k_score<<<sBlocks, wavesPerBlock * 32, 0, stream>>>(
        acc_i, acc_p, q, bq, user, itp, itn, mask, (float*)d_out, B, NU);
}